// GNN_Encoder_80204219286406
// MI455X (gfx1250) — compile-verified
//
#include <hip/hip_runtime.h>

typedef __attribute__((ext_vector_type(2))) float v2f;
typedef __attribute__((ext_vector_type(8))) float v8f;

#define D 64

// ---------------------------------------------------------------------------
// Degree / normalization kernels
// ---------------------------------------------------------------------------
__global__ void k_deg_init(float* __restrict__ deg, int n) {
    int i = blockIdx.x * blockDim.x + threadIdx.x;
    if (i < n) deg[i] = 1.0f;  // self-loop weight
}

__global__ void k_deg_acc(const int* __restrict__ dst, const float* __restrict__ w,
                          float* __restrict__ deg, int E) {
    int e = blockIdx.x * blockDim.x + threadIdx.x;
    if (e < E) atomicAdd(&deg[dst[e]], w[e]);
}

__global__ void k_deg_rsqrt(float* __restrict__ deg, int n) {
    int i = blockIdx.x * blockDim.x + threadIdx.x;
    if (i < n) {
        float d = deg[i];
        deg[i] = (d > 0.0f) ? __frsqrt_rn(d) : 0.0f;
    }
}

// ---------------------------------------------------------------------------
// GEMM: Y[n x 64] = X[n x 64] @ W[64 x 64]  via V_WMMA_F32_16X16X4_F32.
// One wave computes a 16-row x 64-col output slab. n must be a multiple of 16
// (100000 = 6250 * 16). Tail guard is wave-uniform -> EXEC all-1s for WMMA.
// ---------------------------------------------------------------------------
__global__ void k_gemm_wmma(const float* __restrict__ X, const float* __restrict__ W,
                            float* __restrict__ Y, int n) {
    int wave = (int)((blockIdx.x * blockDim.x + threadIdx.x) >> 5);
    int lane = threadIdx.x & 31;
    int m0 = wave * 16;
    if (m0 >= n) return;               // uniform across the wave

    int lhalf = (lane >= 16) ? 1 : 0;  // upper half-wave handles K+2 / M+8
    int lm = lane & 15;

    v8f acc[4] = {v8f{}, v8f{}, v8f{}, v8f{}};

    const float* xrow = X + (size_t)(m0 + lm) * D;

    #pragma unroll
    for (int k0 = 0; k0 < D; k0 += 4) {
        // A fragment (16x4 f32): lane holds row M=lm, K = k0 + 2*lhalf + {0,1}
        v2f a;
        a.x = xrow[k0 + 2 * lhalf + 0];
        a.y = xrow[k0 + 2 * lhalf + 1];
        #pragma unroll
        for (int t = 0; t < 4; ++t) {
            int n0 = t * 16;
            // B fragment (4x16 f32): lane holds col N=n0+lm, K = k0 + 2*lhalf + {0,1}
            v2f b;
            b.x = W[(size_t)(k0 + 2 * lhalf + 0) * D + n0 + lm];
            b.y = W[(size_t)(k0 + 2 * lhalf + 1) * D + n0 + lm];
            acc[t] = __builtin_amdgcn_wmma_f32_16x16x4_f32(
                false, a, false, b, (short)0, acc[t], false, false);
        }
    }

    // C/D layout: VGPR r -> row (r + 8*lhalf), col = lane%16
    #pragma unroll
    for (int t = 0; t < 4; ++t) {
        int col = t * 16 + lm;
        #pragma unroll
        for (int r = 0; r < 8; ++r) {
            int row = m0 + r + 8 * lhalf;
            Y[(size_t)row * D + col] = acc[t][r];
        }
    }
}

// ---------------------------------------------------------------------------
// agg = xl * dinv^2 + bias   (self-loop contribution + conv bias)
// ---------------------------------------------------------------------------
__global__ void k_self_init(const float* __restrict__ xl, const float* __restrict__ dinv,
                            const float* __restrict__ bias, float* __restrict__ agg,
                            int n) {
    size_t tid = (size_t)blockIdx.x * blockDim.x + threadIdx.x;
    if (tid >= (size_t)n * D) return;
    int i = (int)(tid >> 6);
    int f = (int)(tid & 63);
    float di = dinv[i];
    agg[tid] = xl[tid] * di * di + bias[f];
}

// ---------------------------------------------------------------------------
// Edge scatter: one wave per edge; lane handles features lane and lane+32.
// agg[dst] += xl[src] * (dinv[src] * w * dinv[dst])   via global f32 atomics.
// ---------------------------------------------------------------------------
__global__ void k_edge_scatter(const int* __restrict__ src, const int* __restrict__ dst,
                               const float* __restrict__ w, const float* __restrict__ dinv,
                               const float* __restrict__ xl, float* __restrict__ agg,
                               int E) {
    long long gtid = (long long)blockIdx.x * blockDim.x + threadIdx.x;
    int e = (int)(gtid >> 5);
    int lane = (int)(gtid & 31);
    if (e >= E) return;                // uniform across the wave
    int s = src[e];
    int d = dst[e];
    float nrm = dinv[s] * w[e] * dinv[d];
    const float* xs = xl + (size_t)s * D;
    float* ad = agg + (size_t)d * D;
    atomicAdd(&ad[lane],      xs[lane]      * nrm);
    atomicAdd(&ad[lane + 32], xs[lane + 32] * nrm);
}

// ---------------------------------------------------------------------------
// BatchNorm statistics: stats[0:64] = column sums, stats[64:128] = sum of squares
// ---------------------------------------------------------------------------
__global__ void k_zero_stats(float* __restrict__ stats) {
    if (threadIdx.x < 128) stats[threadIdx.x] = 0.0f;
}

__global__ void k_bn_stats(const float* __restrict__ h, float* __restrict__ stats,
                           int n) {
    int f = threadIdx.x & 63;
    int rowsPerBlock = blockDim.x >> 6;
    int row0 = blockIdx.x * rowsPerBlock + (threadIdx.x >> 6);
    int stride = gridDim.x * rowsPerBlock;
    float s = 0.0f, sq = 0.0f;
    for (int r = row0; r < n; r += stride) {
        float v = h[(size_t)r * D + f];
        s += v;
        sq += v * v;
    }
    atomicAdd(&stats[f], s);
    atomicAdd(&stats[D + f], sq);
}

__global__ void k_bn_finalize(const float* __restrict__ stats,
                              const float* __restrict__ gamma,
                              const float* __restrict__ beta,
                              float* __restrict__ ss, float inv_n) {
    int f = threadIdx.x;
    if (f >= D) return;
    float mean = stats[f] * inv_n;
    float var  = stats[D + f] * inv_n - mean * mean;   // biased (jnp.var default)
    float sc = gamma[f] * __frsqrt_rn(var + 1e-5f);
    ss[f]     = sc;
    ss[D + f] = beta[f] - mean * sc;
}

__global__ void k_bn_apply_prelu(const float* __restrict__ h, const float* __restrict__ ss,
                                 const float* __restrict__ alpha, float* __restrict__ out,
                                 int n) {
    size_t tid = (size_t)blockIdx.x * blockDim.x + threadIdx.x;
    if (tid >= (size_t)n * D) return;
    int f = (int)(tid & 63);
    float y = h[tid] * ss[f] + ss[D + f];
    out[tid] = (y >= 0.0f) ? y : alpha[0] * y;
}

// ---------------------------------------------------------------------------
// Host-side launcher
// ---------------------------------------------------------------------------
extern "C" void kernel_launch(void* const* d_in, const int* in_sizes, int n_in,
                              void* d_out, int out_size, void* d_ws, size_t ws_size,
                              hipStream_t stream) {
    const float* x   = (const float*)d_in[0];
    const int*   ei  = (const int*)d_in[1];     // [2, E] flat
    const float* ew  = (const float*)d_in[2];
    const float* W1  = (const float*)d_in[3];
    const float* b1  = (const float*)d_in[4];
    const float* g1  = (const float*)d_in[5];
    const float* be1 = (const float*)d_in[6];
    const float* a1  = (const float*)d_in[7];
    const float* W2  = (const float*)d_in[8];
    const float* b2  = (const float*)d_in[9];
    const float* g2  = (const float*)d_in[10];
    const float* be2 = (const float*)d_in[11];
    const float* a2  = (const float*)d_in[12];

    const int n = in_sizes[0] / D;
    const int E = in_sizes[2];
    const int* srcIdx = ei;
    const int* dstIdx = ei + E;

    // Workspace layout (floats): dinv[n] | bufA[n*D] | bufB[n*D] | stats[128] | ss[128]
    float* ws    = (float*)d_ws;
    float* dinv  = ws;
    float* bufA  = ws + (size_t)((n + 255) & ~255);
    float* bufB  = bufA + (size_t)n * D;
    float* stats = bufB + (size_t)n * D;
    float* ss    = stats + 128;
    float* out   = (float*)d_out;

    const int T = 256;
    const int gN    = (n + T - 1) / T;
    const int gE    = (E + T - 1) / T;
    const int gND   = (int)(((size_t)n * D + T - 1) / T);
    const int gEdge = (int)(((long long)E * 32 + T - 1) / T);
    const int gGemm = (n / 16 + 3) / 4;         // 4 waves (16 rows each) per 128-thread block
    const float inv_n = 1.0f / (float)n;

    // --- normalization coefficients ---
    k_deg_init <<<gN, T, 0, stream>>>(dinv, n);
    k_deg_acc  <<<gE, T, 0, stream>>>(dstIdx, ew, dinv, E);
    k_deg_rsqrt<<<gN, T, 0, stream>>>(dinv, n);

    // ================= Layer 1 =================
    k_gemm_wmma   <<<gGemm, 128, 0, stream>>>(x, W1, bufA, n);           // xl1 = x @ W1
    k_self_init   <<<gND, T, 0, stream>>>(bufA, dinv, b1, bufB, n);      // agg1 init
    k_edge_scatter<<<gEdge, T, 0, stream>>>(srcIdx, dstIdx, ew, dinv, bufA, bufB, E);
    k_zero_stats  <<<1, 128, 0, stream>>>(stats);
    k_bn_stats    <<<512, T, 0, stream>>>(bufB, stats, n);
    k_bn_finalize <<<1, 64, 0, stream>>>(stats, g1, be1, ss, inv_n);
    k_bn_apply_prelu<<<gND, T, 0, stream>>>(bufB, ss, a1, bufA, n);      // h1 -> bufA

    // ================= Layer 2 =================
    k_gemm_wmma   <<<gGemm, 128, 0, stream>>>(bufA, W2, bufB, n);        // xl2 = h1 @ W2
    k_self_init   <<<gND, T, 0, stream>>>(bufB, dinv, b2, bufA, n);      // agg2 init (bufA reused)
    k_edge_scatter<<<gEdge, T, 0, stream>>>(srcIdx, dstIdx, ew, dinv, bufB, bufA, E);
    k_zero_stats  <<<1, 128, 0, stream>>>(stats);
    k_bn_stats    <<<512, T, 0, stream>>>(bufA, stats, n);
    k_bn_finalize <<<1, 64, 0, stream>>>(stats, g2, be2, ss, inv_n);
    k_bn_apply_prelu<<<gND, T, 0, stream>>>(bufA, ss, a2, out, n);       // final output
}